// GetDensity_19301583028811
// MI455X (gfx1250) — compile-verified
//
#include <hip/hip_runtime.h>

#define NWAVE   8
#define NCLASS  118
#define HDIM    64
#define NORBIT  64
#define OC_LOOP 3

typedef __attribute__((ext_vector_type(2))) float v2f;
typedef __attribute__((ext_vector_type(8))) float v8f;

__device__ __forceinline__ float silu_f(float x) { return x / (1.0f + __expf(-x)); }

// ---------------------------------------------------------------------------
// Cooperative global->LDS staging.  Uses CDNA5 async-to-LDS DMA when the
// toolchain exposes it (tracked by ASYNCcnt), else vector loads + stores.
// Callers pass 16B-aligned pointers and n_floats that is a multiple of 4.
// Ends with a full block barrier.
// ---------------------------------------------------------------------------
#if defined(__has_builtin)
#if __has_builtin(__builtin_amdgcn_global_load_async_to_lds_b128) && \
    __has_builtin(__builtin_amdgcn_s_wait_asynccnt)
#define USE_ASYNC_LDS 1
#endif
#endif

typedef int b128_t __attribute__((vector_size(16)));

__device__ __forceinline__ void stage_to_lds(float* lds_dst, const float* gsrc,
                                             int n_floats) {
#if defined(USE_ASYNC_LDS)
  for (int i = threadIdx.x * 4; i < n_floats; i += blockDim.x * 4) {
    __builtin_amdgcn_global_load_async_to_lds_b128(
        (__attribute__((address_space(1))) b128_t*)(gsrc + i),
        (__attribute__((address_space(3))) b128_t*)(lds_dst + i),
        0, 0);
  }
  __builtin_amdgcn_s_wait_asynccnt(0);
#else
  for (int i = threadIdx.x * 4; i < n_floats; i += blockDim.x * 4)
    *(float4*)(lds_dst + i) = *(const float4*)(gsrc + i);
#endif
  __syncthreads();
}

// ---------------------------------------------------------------------------
// One wave computes a 16x16 f32 tile  C += A(16xK) * B(KxN-tile)  via
// V_WMMA_F32_16X16X4_F32, K stepped by 4.  A from global (row-major, lda),
// B from an LDS image of the KxN weight block (row-major, ldb).
// Lane layouts per CDNA5 ISA 7.12.2:
//   A 16x4: lanes 0-15 hold M=lane {K,K+1}; lanes 16-31 hold M=lane-16 {K+2,K+3}
//   B 4x16: vgpr0 = rows K / K+2 across lane halves; vgpr1 = K+1 / K+3
//   D     : vgpr r -> row r (lanes 0-15) and row r+8 (lanes 16-31), col = lane%16
// ---------------------------------------------------------------------------
__device__ __forceinline__ v8f wmma_tile_lds(const float* A, int lda,
                                             const float* Blds, int ldb,
                                             int K, v8f acc) {
  const int lane = threadIdx.x & 31;
  const int half = lane >> 4;
  const int l16  = lane & 15;
  for (int k0 = 0; k0 < K; k0 += 4) {
    const int ka = k0 + 2 * half;
    v2f a = *(const v2f*)(A + l16 * lda + ka);     // contiguous pair -> b64 load
    v2f b;
    b.x = Blds[ka * ldb + l16];
    b.y = Blds[(ka + 1) * ldb + l16];
    acc = __builtin_amdgcn_wmma_f32_16x16x4_f32(false, a, false, b,
                                                (short)0, acc, false, false);
  }
  return acc;
}

// ---------------------------------------------------------------------------
// Species table: ctable[c][m] = MLP_embc(one_hot(c)).   118 blocks x 64 thr.
// ---------------------------------------------------------------------------
__global__ void k_center_table(const float* __restrict__ W1, const float* __restrict__ b1,
                               const float* __restrict__ g,  const float* __restrict__ be,
                               const float* __restrict__ W2, const float* __restrict__ b2,
                               float* __restrict__ ctable) {
  const int c = blockIdx.x, t = threadIdx.x;
  __shared__ float buf[64];
  __shared__ float sact[64];
  float x = W1[c * HDIM + t] + b1[t];
  buf[t] = x; __syncthreads();
  for (int s = 32; s > 0; s >>= 1) { if (t < s) buf[t] += buf[t + s]; __syncthreads(); }
  float mean = buf[0] * (1.0f / 64.0f); __syncthreads();
  float d = x - mean;
  buf[t] = d * d; __syncthreads();
  for (int s = 32; s > 0; s >>= 1) { if (t < s) buf[t] += buf[t + s]; __syncthreads(); }
  float var = buf[0] * (1.0f / 64.0f);
  sact[t] = silu_f(d * rsqrtf(var + 1e-5f) * g[t] + be[t]);
  __syncthreads();
  float acc = b2[t];
  #pragma unroll 8
  for (int k = 0; k < HDIM; ++k) acc += sact[k] * W2[k * NORBIT + t];
  ctable[c * NORBIT + t] = acc;
}

// ---------------------------------------------------------------------------
// Pair table: ptable[a*118+b][0:24] = MLP_embn(one_hot(a)+one_hot(b)).
// ---------------------------------------------------------------------------
__global__ void k_pair_table(const float* __restrict__ W1, const float* __restrict__ b1,
                             const float* __restrict__ g,  const float* __restrict__ be,
                             const float* __restrict__ W2, const float* __restrict__ b2,
                             float* __restrict__ ptable) {
  const int pc = blockIdx.x, t = threadIdx.x;
  const int a = pc / NCLASS, b = pc % NCLASS;
  __shared__ float buf[64];
  __shared__ float sact[64];
  float x = W1[a * HDIM + t] + W1[b * HDIM + t] + b1[t];
  buf[t] = x; __syncthreads();
  for (int s = 32; s > 0; s >>= 1) { if (t < s) buf[t] += buf[t + s]; __syncthreads(); }
  float mean = buf[0] * (1.0f / 64.0f); __syncthreads();
  float d = x - mean;
  buf[t] = d * d; __syncthreads();
  for (int s = 32; s > 0; s >>= 1) { if (t < s) buf[t] += buf[t + s]; __syncthreads(); }
  float var = buf[0] * (1.0f / 64.0f);
  sact[t] = silu_f(d * rsqrtf(var + 1e-5f) * g[t] + be[t]);
  __syncthreads();
  if (t < 3 * NWAVE) {
    float acc = b2[t];
    #pragma unroll 8
    for (int k = 0; k < HDIM; ++k) acc += sact[k] * W2[k * (3 * NWAVE) + t];
    ptable[pc * (3 * NWAVE) + t] = acc;
  }
}

// ---------------------------------------------------------------------------
// Per-pair geometry: dist_vec (to output), ang[4], radial[8], iter_coeff[8].
// ---------------------------------------------------------------------------
__global__ void k_pairs(const float* __restrict__ cart, const int* __restrict__ ai,
                        const int* __restrict__ species, const float* __restrict__ ptable,
                        float* __restrict__ angrad, float* __restrict__ iterc,
                        float* __restrict__ distvec, int npairs) {
  const int p = blockIdx.x * blockDim.x + threadIdx.x;
  if (p >= npairs) return;
  const int i0 = ai[p], i1 = ai[npairs + p];
  const float dx = cart[i0 * 3 + 0] - cart[i1 * 3 + 0];
  const float dy = cart[i0 * 3 + 1] - cart[i1 * 3 + 1];
  const float dz = cart[i0 * 3 + 2] - cart[i1 * 3 + 2];
  distvec[p * 3 + 0] = dx; distvec[p * 3 + 1] = dy; distvec[p * 3 + 2] = dz;
  const float d = sqrtf(dx * dx + dy * dy + dz * dz);
  const int s0 = species[i0] + 1, s1 = species[i1] + 1;
  const float* e = ptable + (size_t)(s0 * NCLASS + s1) * (3 * NWAVE);
  float fc = 0.5f * __cosf(d * 0.78539816339744830961f) + 0.5f;   // pi/CUTOFF = pi/4
  fc = fc * fc;
  float* ar = angrad + (size_t)p * 12;
  ar[0] = fc; ar[1] = dx * fc; ar[2] = dy * fc; ar[3] = dz * fc;
  #pragma unroll
  for (int k = 0; k < NWAVE; ++k) {
    const float t = e[NWAVE + k] * (d - e[2 * NWAVE + k]);
    ar[4 + k] = __expf(-t * t);
    iterc[(size_t)p * NWAVE + k] = e[k];
  }
}

// ---------------------------------------------------------------------------
// Scatter: center_orbital[i0][j*8+k] += ang[j]*radial[k]*iter_coeff[k]
// ---------------------------------------------------------------------------
__global__ void k_scatter(const float* __restrict__ angrad, const float* __restrict__ iterc,
                          const int* __restrict__ ai, float* __restrict__ co, int npairs) {
  const int p = blockIdx.x * blockDim.x + threadIdx.x;
  if (p >= npairs) return;
  if (p + 4096 < npairs) {                 // stream prefetch (global_prefetch_b8)
    __builtin_prefetch(angrad + (size_t)(p + 4096) * 12, 0, 1);
    __builtin_prefetch(iterc + (size_t)(p + 4096) * NWAVE, 0, 1);
  }
  const int i0 = ai[p];
  const float* ar = angrad + (size_t)p * 12;
  const float a0 = ar[0], a1 = ar[1], a2 = ar[2], a3 = ar[3];
  float* dst = co + (size_t)i0 * 32;
  #pragma unroll
  for (int k = 0; k < NWAVE; ++k) {
    const float w = ar[4 + k] * iterc[(size_t)p * NWAVE + k];
    atomicAdd(dst +       k, a0 * w);
    atomicAdd(dst +  8 +  k, a1 * w);
    atomicAdd(dst + 16 +  k, a2 * w);
    atomicAdd(dst + 24 +  k, a3 * w);
  }
}

// ---------------------------------------------------------------------------
// density[i][m] = sum_j (sum_k co[i][j*8+k] * cc[l][para_j][k][m])^2 + ctable
// Block = 128 threads (4 waves); each wave owns one 16-col n-tile of the same
// 16-row m-tile.  The 2x(8x64) coefficient block is staged to LDS once.
// ---------------------------------------------------------------------------
__global__ void k_density(const float* __restrict__ co, const float* __restrict__ cc,
                          const int* __restrict__ species, const float* __restrict__ ctable,
                          float* __restrict__ density, int l) {
  __shared__ __attribute__((aligned(16))) float ccs[2 * 8 * NORBIT];   // 4 KB
  stage_to_lds(ccs, cc + (size_t)(l * 2) * 8 * NORBIT, 2 * 8 * NORBIT);

  const int i0 = blockIdx.x * 16;
  const int wave = threadIdx.x >> 5;
  const int n0 = wave * 16;
  v8f dacc = {};
  #pragma unroll
  for (int j = 0; j < 4; ++j) {
    const float* A = co + (size_t)i0 * 32 + j * 8;      // 16 x 8, lda 32
    const int para = (j > 0) ? 1 : 0;                   // INDEX_PARA = [0,1,1,1]
    const float* B = ccs + para * 8 * NORBIT + n0;      // 8 x 64 slice in LDS
    v8f acc = {};
    acc = wmma_tile_lds(A, 32, B, NORBIT, 8, acc);
    #pragma unroll
    for (int r = 0; r < 8; ++r) dacc[r] += acc[r] * acc[r];
  }
  const int lane = threadIdx.x & 31, half = lane >> 4, l16 = lane & 15;
  #pragma unroll
  for (int r = 0; r < 8; ++r) {
    const int row = i0 + r + 8 * half;
    const int m = n0 + l16;
    density[(size_t)row * NORBIT + m] = dacc[r] + ctable[(species[row] + 1) * NORBIT + m];
  }
}

// ---------------------------------------------------------------------------
// Y(N,64) = X(N,64) @ W(64,64) + bias.  Block = 128 (4 waves = 4 n-tiles of
// one m-tile); W staged to LDS once per block.  grid = N/16.
// ---------------------------------------------------------------------------
__global__ void k_gemm64(const float* __restrict__ X, const float* __restrict__ W,
                         const float* __restrict__ bias, float* __restrict__ Y) {
  __shared__ __attribute__((aligned(16))) float Ws[HDIM * HDIM];       // 16 KB
  stage_to_lds(Ws, W, HDIM * HDIM);

  const int i0 = blockIdx.x * 16;
  const int wave = threadIdx.x >> 5;
  const int n0 = wave * 16;
  v8f acc = {};
  acc = wmma_tile_lds(X + (size_t)i0 * HDIM, HDIM, Ws + n0, HDIM, HDIM, acc);
  const int lane = threadIdx.x & 31, half = lane >> 4, l16 = lane & 15;
  #pragma unroll
  for (int r = 0; r < 8; ++r)
    Y[(size_t)(i0 + r + 8 * half) * HDIM + n0 + l16] = acc[r] + bias[n0 + l16];
}

// ---------------------------------------------------------------------------
// Y(N,8) = X(N,64) @ W(64,8) + bias.  N-tile padded to 16 with zero columns.
// grid = N/16, block = 32.
// ---------------------------------------------------------------------------
__global__ void k_gemm64x8(const float* __restrict__ X, const float* __restrict__ W,
                           const float* __restrict__ bias, float* __restrict__ Y) {
  const int i0 = blockIdx.x * 16;
  const int lane = threadIdx.x & 31, half = lane >> 4, l16 = lane & 15;
  const float* A = X + (size_t)i0 * HDIM;
  v8f acc = {};
  for (int k0 = 0; k0 < HDIM; k0 += 4) {
    const int ka = k0 + 2 * half;
    v2f a = *(const v2f*)(A + l16 * HDIM + ka);
    v2f b;
    b.x = (l16 < NWAVE) ? W[ka * NWAVE + l16] : 0.0f;
    b.y = (l16 < NWAVE) ? W[(ka + 1) * NWAVE + l16] : 0.0f;
    acc = __builtin_amdgcn_wmma_f32_16x16x4_f32(false, a, false, b,
                                                (short)0, acc, false, false);
  }
  if (l16 < NWAVE) {
    #pragma unroll
    for (int r = 0; r < 8; ++r)
      Y[(size_t)(i0 + r + 8 * half) * NWAVE + l16] = acc[r] + bias[l16];
  }
}

// ---------------------------------------------------------------------------
// LayerNorm + SiLU over rows of 64.  grid = natoms, block = 64.
// ---------------------------------------------------------------------------
__global__ void k_lnact(const float* __restrict__ pre, const float* __restrict__ g,
                        const float* __restrict__ be, float* __restrict__ h) {
  const int i = blockIdx.x, t = threadIdx.x;
  __shared__ float buf[64];
  float x = pre[(size_t)i * HDIM + t];
  buf[t] = x; __syncthreads();
  for (int s = 32; s > 0; s >>= 1) { if (t < s) buf[t] += buf[t + s]; __syncthreads(); }
  float mean = buf[0] * (1.0f / 64.0f); __syncthreads();
  float d = x - mean;
  buf[t] = d * d; __syncthreads();
  for (int s = 32; s > 0; s >>= 1) { if (t < s) buf[t] += buf[t + s]; __syncthreads(); }
  float var = buf[0] * (1.0f / 64.0f);
  h[(size_t)i * HDIM + t] = silu_f(d * rsqrtf(var + 1e-5f) * g[t] + be[t]);
}

// ---------------------------------------------------------------------------
// iter_coeff[p][k] += nnout[atom_index[1][p]][k]
// ---------------------------------------------------------------------------
__global__ void k_update(const int* __restrict__ ai, const float* __restrict__ nnout,
                         float* __restrict__ iterc, int npairs) {
  const int idx = blockIdx.x * blockDim.x + threadIdx.x;
  const int p = idx >> 3, k = idx & 7;
  if (p >= npairs) return;
  const int i1 = ai[npairs + p];
  iterc[(size_t)p * NWAVE + k] += nnout[(size_t)i1 * NWAVE + k];
}

// ---------------------------------------------------------------------------
// Final head: LN + SiLU + dot(out_W2) + bias, masked.  grid = natoms, block = 64.
// ---------------------------------------------------------------------------
__global__ void k_out(const float* __restrict__ pre, const float* __restrict__ g,
                      const float* __restrict__ be, const float* __restrict__ W2,
                      const float* __restrict__ b2, const int* __restrict__ species,
                      float* __restrict__ out) {
  const int i = blockIdx.x, t = threadIdx.x;
  __shared__ float buf[64];
  float x = pre[(size_t)i * HDIM + t];
  buf[t] = x; __syncthreads();
  for (int s = 32; s > 0; s >>= 1) { if (t < s) buf[t] += buf[t + s]; __syncthreads(); }
  float mean = buf[0] * (1.0f / 64.0f); __syncthreads();
  float d = x - mean;
  buf[t] = d * d; __syncthreads();
  for (int s = 32; s > 0; s >>= 1) { if (t < s) buf[t] += buf[t + s]; __syncthreads(); }
  float var = buf[0] * (1.0f / 64.0f);
  __syncthreads();
  buf[t] = silu_f(d * rsqrtf(var + 1e-5f) * g[t] + be[t]) * W2[t];
  __syncthreads();
  for (int s = 32; s > 0; s >>= 1) { if (t < s) buf[t] += buf[t + s]; __syncthreads(); }
  if (t == 0) {
    const float mask = (species[i] >= 0) ? 1.0f : 0.0f;   // species > -0.5
    out[i] = (buf[0] + b2[0]) * mask;
  }
}

// ---------------------------------------------------------------------------
extern "C" void kernel_launch(void* const* d_in, const int* in_sizes, int n_in,
                              void* d_out, int out_size, void* d_ws, size_t ws_size,
                              hipStream_t stream) {
  const float* cart     = (const float*)d_in[0];
  const int*   ai       = (const int*)  d_in[1];
  const int*   species  = (const int*)  d_in[2];
  /* d_in[3] neigh_species unused */
  const float* cc       = (const float*)d_in[4];
  const float* embn_W1  = (const float*)d_in[5];
  const float* embn_b1  = (const float*)d_in[6];
  const float* embn_g   = (const float*)d_in[7];
  const float* embn_be  = (const float*)d_in[8];
  const float* embn_W2  = (const float*)d_in[9];
  const float* embn_b2  = (const float*)d_in[10];
  const float* embc_W1  = (const float*)d_in[11];
  const float* embc_b1  = (const float*)d_in[12];
  const float* embc_g   = (const float*)d_in[13];
  const float* embc_be  = (const float*)d_in[14];
  const float* embc_W2  = (const float*)d_in[15];
  const float* embc_b2  = (const float*)d_in[16];
  const float* oc_W1    = (const float*)d_in[17];
  const float* oc_b1    = (const float*)d_in[18];
  const float* oc_g     = (const float*)d_in[19];
  const float* oc_be    = (const float*)d_in[20];
  const float* oc_W2    = (const float*)d_in[21];
  const float* oc_b2    = (const float*)d_in[22];
  const float* out_W1   = (const float*)d_in[23];
  const float* out_b1   = (const float*)d_in[24];
  const float* out_g    = (const float*)d_in[25];
  const float* out_be   = (const float*)d_in[26];
  const float* out_W2   = (const float*)d_in[27];
  const float* out_b2   = (const float*)d_in[28];

  const int npairs = in_sizes[1] / 2;
  const int natoms = in_sizes[2];

  float* ws = (float*)d_ws;
  size_t off = 0;
  float* angrad  = ws + off; off += (size_t)npairs * 12;
  float* iterc   = ws + off; off += (size_t)npairs * NWAVE;
  float* co      = ws + off; off += (size_t)natoms * 32;
  float* density = ws + off; off += (size_t)natoms * NORBIT;
  float* pre     = ws + off; off += (size_t)natoms * HDIM;
  float* hbuf    = ws + off; off += (size_t)natoms * HDIM;
  float* nnout   = ws + off; off += (size_t)natoms * NWAVE;
  float* ctable  = ws + off; off += (size_t)NCLASS * NORBIT;
  float* ptable  = ws + off; off += (size_t)NCLASS * NCLASS * (3 * NWAVE);

  float* distvec = (float*)d_out;
  float* energy  = (float*)d_out + (size_t)npairs * 3;

  const int pairBlocks = (npairs + 255) / 256;
  const int mtiles = natoms / 16;

  k_center_table<<<NCLASS, 64, 0, stream>>>(embc_W1, embc_b1, embc_g, embc_be,
                                            embc_W2, embc_b2, ctable);
  k_pair_table<<<NCLASS * NCLASS, 64, 0, stream>>>(embn_W1, embn_b1, embn_g, embn_be,
                                                   embn_W2, embn_b2, ptable);
  k_pairs<<<pairBlocks, 256, 0, stream>>>(cart, ai, species, ptable,
                                          angrad, iterc, distvec, npairs);

  for (int l = 0; l <= OC_LOOP; ++l) {
    (void)hipMemsetAsync(co, 0, (size_t)natoms * 32 * sizeof(float), stream);
    k_scatter<<<pairBlocks, 256, 0, stream>>>(angrad, iterc, ai, co, npairs);
    k_density<<<mtiles, 128, 0, stream>>>(co, cc, species, ctable, density, l);
    if (l < OC_LOOP) {
      k_gemm64<<<mtiles, 128, 0, stream>>>(density, oc_W1 + (size_t)l * HDIM * HDIM,
                                           oc_b1 + l * HDIM, pre);
      k_lnact<<<natoms, 64, 0, stream>>>(pre, oc_g + l * HDIM, oc_be + l * HDIM, hbuf);
      k_gemm64x8<<<mtiles, 32, 0, stream>>>(hbuf, oc_W2 + (size_t)l * HDIM * NWAVE,
                                            oc_b2 + l * NWAVE, nnout);
      k_update<<<(npairs * 8 + 255) / 256, 256, 0, stream>>>(ai, nnout, iterc, npairs);
    }
  }

  k_gemm64<<<mtiles, 128, 0, stream>>>(density, out_W1, out_b1, pre);
  k_out<<<natoms, 64, 0, stream>>>(pre, out_g, out_be, out_W2, out_b2, species, energy);
}